// EMD_loss_53128745451533
// MI455X (gfx1250) — compile-verified
//
#include <hip/hip_runtime.h>
#include <math.h>

#define N_BINS 360
#define BATCH 65536
#define ROWS_PER_TILE 16
#define K4 (N_BINS / 4)          /* 90 float4 chunks per row / 90 k-steps */
#define LDS_STRIDE 364           /* padded row stride (floats): conflict-free fragment reads */
#define NBLOCKS 2048
#define NTILES (BATCH / ROWS_PER_TILE)  /* 4096 */

typedef __attribute__((ext_vector_type(2))) float v2f;
typedef __attribute__((ext_vector_type(8))) float v8f;

__global__ __launch_bounds__(256) void emd_partial_kernel(
    const float* __restrict__ y_pred,
    const int*   __restrict__ y_labels,
    float*       __restrict__ partials)
{
    __shared__ float lds[ROWS_PER_TILE * LDS_STRIDE];
    __shared__ float wsum[8];

    const int tid    = threadIdx.x;
    const int lane   = tid & 31;
    const int waveId = tid >> 5;      // 8 wave32s per block
    const int m      = lane & 15;     // A-fragment row (M)
    const int half   = lane >> 4;     // 0: K=c0,c0+1 ; 1: K=c0+2,c0+3

    const float norm   = 0.07978845608028654f;  // 1/(sqrt(2*pi)*5)
    const float inv2s2 = 0.02f;                 // 1/(2*sigma^2)

    v8f acc = {0.f,0.f,0.f,0.f,0.f,0.f,0.f,0.f};
    v2f bones; bones.x = 1.0f; bones.y = 1.0f;  // B = ones -> D = rowsum(A) + C

    for (int tile = blockIdx.x; tile < NTILES; tile += gridDim.x) {
        const int row0 = tile * ROWS_PER_TILE;

        // ---- Phase 1: async DMA raw y_pred tile -> LDS (no VGPR staging) ----
        for (int f = tid; f < ROWS_PER_TILE * K4; f += 256) {
            const int r  = f / K4;
            const int c4 = f - r * K4;
            const float* gsrc = y_pred + (size_t)(row0 + r) * N_BINS + c4 * 4;
            const unsigned ldsOff =
                (unsigned)(size_t)(&lds[r * LDS_STRIDE + c4 * 4]);   // byte offset in LDS
            asm volatile("global_load_async_to_lds_b128 %0, %1, off"
                         :: "v"(ldsOff), "v"((unsigned long long)(size_t)gsrc)
                         : "memory");
        }
        asm volatile("s_wait_asynccnt 0" ::: "memory");
        __syncthreads();

        // ---- Phase 2: LDS fragment read -> Gaussian + sq-diff -> WMMA row-sum ----
        const float flbl = (float)y_labels[row0 + m];
        for (int step = waveId; step < K4; step += 8) {      // wave-uniform -> EXEC all 1s
            const int base = m * LDS_STRIDE + step * 4 + half * 2;   // 8B-aligned pair
            const float p0 = lds[base];
            const float p1 = lds[base + 1];
            const float c0 = (float)(step * 4 + half * 2);
            const float d0 = c0 - flbl;
            const float d1 = (c0 + 1.0f) - flbl;
            const float e0 = __expf(-(d0 * d0) * inv2s2) * norm - p0;
            const float e1 = __expf(-(d1 * d1) * inv2s2) * norm - p1;
            v2f a;
            a.x = e0 * e0;
            a.y = e1 * e1;
            acc = __builtin_amdgcn_wmma_f32_16x16x4_f32(
                /*neg_a=*/false, a, /*neg_b=*/false, bones,
                /*c_mod=*/(short)0, acc, /*reuse_a=*/false, /*reuse_b=*/false);
        }
        __syncthreads();   // protect LDS before next tile's DMA overwrites it
    }

    // ---- drain accumulator: every D column holds the same row-sum (B was ones) ----
    float local = acc[0] + acc[1] + acc[2] + acc[3] + acc[4] + acc[5] + acc[6] + acc[7];
    for (int off = 16; off > 0; off >>= 1)
        local += __shfl_xor(local, off, 32);
    if (lane == 0) wsum[waveId] = local * (1.0f / 16.0f);    // undo 16x column replication
    __syncthreads();
    if (tid == 0) {
        float s = 0.f;
        for (int i = 0; i < 8; ++i) s += wsum[i];
        partials[blockIdx.x] = s;                            // written unconditionally
    }
}

__global__ __launch_bounds__(256) void emd_final_kernel(
    const float* __restrict__ partials,
    float*       __restrict__ out,
    int n)
{
    __shared__ double red[256];
    double s = 0.0;
    for (int i = threadIdx.x; i < n; i += 256) s += (double)partials[i];
    red[threadIdx.x] = s;
    __syncthreads();
    for (int off = 128; off > 0; off >>= 1) {
        if ((int)threadIdx.x < off) red[threadIdx.x] += red[threadIdx.x + off];
        __syncthreads();
    }
    if (threadIdx.x == 0) out[0] = (float)(red[0] * (1.0 / (double)BATCH));
}

extern "C" void kernel_launch(void* const* d_in, const int* in_sizes, int n_in,
                              void* d_out, int out_size, void* d_ws, size_t ws_size,
                              hipStream_t stream)
{
    const float* y_pred   = (const float*)d_in[0];
    const int*   y_labels = (const int*)d_in[1];
    float*       out      = (float*)d_out;
    float*       partials = (float*)d_ws;   // NBLOCKS * 4 bytes

    emd_partial_kernel<<<NBLOCKS, 256, 0, stream>>>(y_pred, y_labels, partials);
    emd_final_kernel<<<1, 256, 0, stream>>>(partials, out, NBLOCKS);
}